// BVHQwenRouterAdapter_49323404427406
// MI455X (gfx1250) — compile-verified
//
#include <hip/hip_runtime.h>

#define T_TOKENS 16384
#define HID      2048
#define NEXP     128
#define TOPK     8
#define NCAND    32
#define WAVES    2          // waves (16-token tiles) per workgroup

typedef __attribute__((ext_vector_type(2))) float v2f;
typedef __attribute__((ext_vector_type(4))) float v4f;
typedef __attribute__((ext_vector_type(8))) float v8f;

#define NEG_INF (-1e30f)

// ---------------------------------------------------------------------------
// Prepass: repack weight matrix [128,2048] row-major into WMMA-B-fragment
// order so the GEMM hot loop issues fully coalesced b128 loads.
// Packed index: ((n*128 + hc)*32 + lane)*8 + je  with je = 2*j + v,
// source element W[16n + lane%16][16*hc + 4*j + 2*(lane/16) + v]
// (B 4x16 fp32 layout: N = lane%16, K = v + 2*(lane/16)).
// ---------------------------------------------------------------------------
__global__ __launch_bounds__(256) void pack_weights_kernel(
    const float* __restrict__ W, float* __restrict__ P) {
  int tid  = blockIdx.x * 256 + threadIdx.x;     // 0 .. 262143
  int je   = tid & 7;
  int lane = (tid >> 3) & 31;
  int hc   = (tid >> 8) & 127;
  int n    = tid >> 15;
  int e = n * 16 + (lane & 15);
  int h = hc * 16 + ((je >> 1) << 2) + ((lane >> 4) << 1) + (je & 1);
  P[tid] = W[(size_t)e * HID + h];
}

// ---------------------------------------------------------------------------
// Fused router: dual fp32 WMMA GEMM (16 tokens x 128 experts x 2048) +
// softmax + bvh top-32 candidate prune + top-8 select + renorm.
// ---------------------------------------------------------------------------
__global__ __launch_bounds__(WAVES * 32) void router_kernel(
    const float* __restrict__ x,
    const float* __restrict__ Po,   // packed original_weight
    const float* __restrict__ Pb,   // packed bvh_weight
    float* __restrict__ out_full,   // [T,128]
    float* __restrict__ out_vals,   // [T,8]
    float* __restrict__ out_idx) {  // [T,8] (indices as floats)
  const int lane = threadIdx.x & 31;
  const int wave = threadIdx.x >> 5;
  const int tb   = (blockIdx.x * WAVES + wave) * 16;   // token tile base
  const int koff = (lane >> 4) << 1;                   // A-frag K offset
  const float* xrow = x + (size_t)(tb + (lane & 15)) * HID;

  v8f acc_o[8], acc_b[8];
#pragma unroll
  for (int n = 0; n < 8; ++n) { acc_o[n] = v8f{}; acc_b[n] = v8f{}; }

  const size_t lanebase = (size_t)lane * 8;
  // -------- GEMM main loop: 128 chunks of K=16, 64 WMMAs per chunk --------
  for (int hc = 0; hc < HID / 16; ++hc) {
    v2f a[4];
    const float* ab = xrow + hc * 16 + koff;
#pragma unroll
    for (int j = 0; j < 4; ++j) a[j] = *(const v2f*)(ab + 4 * j);

    const size_t base = (size_t)hc * 256 + lanebase;
#pragma unroll
    for (int n = 0; n < 8; ++n) {
      const float* pon = Po + base + (size_t)n * 32768;
      const float* pbn = Pb + base + (size_t)n * 32768;
      v4f b0 = *(const v4f*)(pon);
      v4f b1 = *(const v4f*)(pon + 4);
      v4f c0 = *(const v4f*)(pbn);
      v4f c1 = *(const v4f*)(pbn + 4);
      v2f f;
#define WM(ACC, AJ, E0, E1)                                                  \
      f[0] = (E0); f[1] = (E1);                                              \
      ACC = __builtin_amdgcn_wmma_f32_16x16x4_f32(                           \
          false, (AJ), false, f, (short)0, ACC, false, false)
      WM(acc_o[n], a[0], b0[0], b0[1]);
      WM(acc_o[n], a[1], b0[2], b0[3]);
      WM(acc_o[n], a[2], b1[0], b1[1]);
      WM(acc_o[n], a[3], b1[2], b1[3]);
      WM(acc_b[n], a[0], c0[0], c0[1]);
      WM(acc_b[n], a[1], c0[2], c0[3]);
      WM(acc_b[n], a[2], c1[0], c1[1]);
      WM(acc_b[n], a[3], c1[2], c1[3]);
#undef WM
    }
  }

  // -------- spill logits to LDS (per-wave region, no cross-wave sync) -----
  __shared__ float lds_full[WAVES][16][NEXP];
  __shared__ float lds_bvh [WAVES][16][NEXP];
  {
    const int mh = (lane >> 4) << 3;   // C/D layout: M = r + 8*(lane/16)
    const int eb = lane & 15;          //             N = lane%16
#pragma unroll
    for (int n = 0; n < 8; ++n)
#pragma unroll
      for (int r = 0; r < 8; ++r) {
        lds_full[wave][mh + r][n * 16 + eb] = acc_o[n][r];
        lds_bvh [wave][mh + r][n * 16 + eb] = acc_b[n][r];
      }
  }

  // -------- per-token epilogue: softmax + candidate prune + top-8 ---------
  for (int t = 0; t < 16; ++t) {
    float lf[4], lb[4];
#pragma unroll
    for (int q = 0; q < 4; ++q) {
      lf[q] = lds_full[wave][t][lane + 32 * q];
      lb[q] = lds_bvh [wave][t][lane + 32 * q];
    }
    // softmax over 128 experts (exact router)
    float mx = fmaxf(fmaxf(lf[0], lf[1]), fmaxf(lf[2], lf[3]));
#pragma unroll
    for (int off = 16; off > 0; off >>= 1)
      mx = fmaxf(mx, __shfl_xor(mx, off, 32));
    float p[4], s = 0.0f;
#pragma unroll
    for (int q = 0; q < 4; ++q) { p[q] = __expf(lf[q] - mx); s += p[q]; }
#pragma unroll
    for (int off = 16; off > 0; off >>= 1) s += __shfl_xor(s, off, 32);
    const float rs = __builtin_amdgcn_rcpf(s);
#pragma unroll
    for (int q = 0; q < 4; ++q) {
      p[q] *= rs;
      out_full[(size_t)(tb + t) * NEXP + lane + 32 * q] = p[q];
    }

    // stage 1: top-32 experts by bvh logit (softmax is monotone).
    // candidate c lands in lane c as (expert id, exact prob).
    unsigned avail = 0xFu;
    int   cand_e = 0;
    float cand_p = NEG_INF;
    for (int c = 0; c < NCAND; ++c) {
      float bv = NEG_INF;
      int   bi = lane;
#pragma unroll
      for (int q = 0; q < 4; ++q) {
        float vq = ((avail >> q) & 1u) ? lb[q] : NEG_INF;
        int   eq = lane + 32 * q;
        if (vq > bv) { bv = vq; bi = eq; }
      }
#pragma unroll
      for (int off = 16; off > 0; off >>= 1) {
        float ov = __shfl_xor(bv, off, 32);
        int   oi = __shfl_xor(bi, off, 32);
        if (ov > bv || (ov == bv && oi < bi)) { bv = ov; bi = oi; }
      }
      if (lane == (bi & 31)) avail &= ~(1u << (bi >> 5));
      int qsel = bi >> 5;   // wave-uniform
      float pl = qsel == 0 ? p[0] : qsel == 1 ? p[1] : qsel == 2 ? p[2] : p[3];
      float pv = __shfl(pl, bi & 31, 32);
      if (lane == c) { cand_e = bi; cand_p = pv; }
    }

    // stage 2: top-8 among candidates by exact prob, tie-break by candidate
    // position (== reference's gather order), then renormalize.
    float cp = cand_p;
    float sum8 = 0.0f, myv = 0.0f;
    int   mye = 0;
    for (int k = 0; k < TOPK; ++k) {
      float v = cp;
      int   i = lane;
#pragma unroll
      for (int off = 16; off > 0; off >>= 1) {
        float ov = __shfl_xor(v, off, 32);
        int   oi = __shfl_xor(i, off, 32);
        if (ov > v || (ov == v && oi < i)) { v = ov; i = oi; }
      }
      int we = __shfl(cand_e, i, 32);
      sum8 += v;
      if (lane == k) { myv = v; mye = we; }
      if (lane == i) cp = NEG_INF;
    }
    const float rn = __builtin_amdgcn_rcpf(sum8);
    if (lane < TOPK) {
      out_vals[(size_t)(tb + t) * TOPK + lane] = myv * rn;
      out_idx [(size_t)(tb + t) * TOPK + lane] = (float)mye;
    }
  }
}

extern "C" void kernel_launch(void* const* d_in, const int* in_sizes, int n_in,
                              void* d_out, int out_size, void* d_ws, size_t ws_size,
                              hipStream_t stream) {
  (void)in_sizes; (void)n_in; (void)out_size; (void)ws_size;
  const float* x  = (const float*)d_in[0];   // hidden_states [T,H]
  const float* Wo = (const float*)d_in[1];   // original_weight [E,H]
  const float* Wb = (const float*)d_in[2];   // bvh_weight [E,H]

  float* Po = (float*)d_ws;                          // 1 MB
  float* Pb = Po + (size_t)NEXP * HID;               // 1 MB

  float* out_full = (float*)d_out;                   // [T,128]
  float* out_vals = out_full + (size_t)T_TOKENS * NEXP;
  float* out_idx  = out_vals + (size_t)T_TOKENS * TOPK;

  pack_weights_kernel<<<dim3((NEXP * HID) / 256), dim3(256), 0, stream>>>(Wo, Po);
  pack_weights_kernel<<<dim3((NEXP * HID) / 256), dim3(256), 0, stream>>>(Wb, Pb);

  router_kernel<<<dim3(T_TOKENS / (16 * WAVES)), dim3(32 * WAVES), 0, stream>>>(
      x, Po, Pb, out_full, out_vals, out_idx);
}